// HypersphericalProto_24610162606420
// MI455X (gfx1250) — compile-verified
//
#include <hip/hip_runtime.h>
#include <math.h>

typedef __attribute__((ext_vector_type(2))) float v2f;
typedef __attribute__((ext_vector_type(4))) float v4f;
typedef __attribute__((ext_vector_type(8))) float v8f;

#define EPS 1e-8f

// Problem geometry (matches reference setup_inputs)
static constexpr int C_ = 128;   // embedding dim (reduction)
static constexpr int K_ = 512;   // number of prototypes (output cols)

// Block tile
static constexpr int BM  = 64;   // output rows per block
static constexpr int BN  = 64;   // output cols per block
static constexpr int LDA = 132;  // padded LDS stride (floats) for A tile: banks (4m+k)%64 conflict-free
static constexpr int LDB = 80;   // padded LDS stride (floats) for B tile: 2*80 % 64 == 32 -> lane halves hit disjoint quadrants

// ---------------------------------------------------------------------------
// Kernel 1: normalize prototypes and store transposed: pT[c*K_ + n] = p[n][c]/||p[n]||
// K_ blocks x C_ threads. Tiny (256 KB out), stays hot in L2 for the GEMM.
// ---------------------------------------------------------------------------
__global__ void proto_norm_T_kernel(const float* __restrict__ proto,
                                    float* __restrict__ pT) {
    __shared__ float red[C_];
    const int row = blockIdx.x;    // prototype index 0..K_-1
    const int t   = threadIdx.x;   // 0..C_-1
    const float v = proto[(size_t)row * C_ + t];
    red[t] = v * v;
    __syncthreads();
#pragma unroll
    for (int s = C_ / 2; s > 0; s >>= 1) {
        if (t < s) red[t] += red[t + s];
        __syncthreads();
    }
    const float inv = 1.0f / fmaxf(sqrtf(red[0]), EPS);
    pT[(size_t)t * K_ + row] = v * inv;
}

// ---------------------------------------------------------------------------
// Kernel 2: fused normalize(embeddings) + GEMM(e_n, p_n^T) + score epilogue.
// grid = (N/BM, K_/BN), block = 256 threads (8 waves, wave32).
// Wave w -> n-tile (w&3), m-tiles {2*(w>>2), 2*(w>>2)+1}; B frag reused across both.
// ---------------------------------------------------------------------------
__global__ __launch_bounds__(256, 2)
void cos_score_gemm_kernel(const float* __restrict__ emb,
                           const float* __restrict__ pT,
                           float* __restrict__ out) {
    __shared__ float As[BM * LDA];        // 64 x 132 f32  (33.8 KB)
    __shared__ float Bs[C_ * LDB];        // 128 x 80 f32  (40.0 KB)
    __shared__ float partial[BM][4];
    __shared__ float rinv[BM];

    const int tid     = threadIdx.x;
    const int rowBase = blockIdx.x * BM;  // along N
    const int colBase = blockIdx.y * BN;  // along K_

    // ---- Stage A tile (64x128) + per-thread partial sum-of-squares.
    // Thread t loads 32 contiguous floats: row = t>>2, segment = t&3.
    {
        const int r   = tid >> 2;
        const int seg = tid & 3;
        const float* src = emb + (size_t)(rowBase + r) * C_ + seg * 32;
        float*       dst = &As[r * LDA + seg * 32];
        float ss = 0.0f;
#pragma unroll
        for (int i = 0; i < 8; ++i) {
            v4f x = *(const v4f*)(src + 4 * i);
            *(v4f*)(dst + 4 * i) = x;
            ss += x.x * x.x + x.y * x.y + x.z * x.z + x.w * x.w;
        }
        partial[r][seg] = ss;
    }
    // ---- Stage B tile (128x64) from pT (row stride K_), coalesced.
    // Thread t: 8 consecutive float4 -> half a row. row = t>>1, half = t&1.
    {
        const int br  = tid >> 1;
        const int bh  = (tid & 1) * 32;  // float offset within the 64-col tile
        const float* src = pT + (size_t)br * K_ + colBase + bh;
        float*       dst = &Bs[br * LDB + bh];
#pragma unroll
        for (int i = 0; i < 8; ++i)
            *(v4f*)(dst + 4 * i) = *(const v4f*)(src + 4 * i);
    }
    __syncthreads();

    // ---- Per-row 1/max(||e||, eps)
    if (tid < BM) {
        const float s = partial[tid][0] + partial[tid][1] +
                        partial[tid][2] + partial[tid][3];
        rinv[tid] = 1.0f / fmaxf(sqrtf(s), EPS);
    }
    __syncthreads();

    // ---- Scale A tile in place (same mapping as the staging pass).
    {
        const int r   = tid >> 2;
        const int seg = tid & 3;
        const float s = rinv[r];
        float* dst = &As[r * LDA + seg * 32];
#pragma unroll
        for (int i = 0; i < 8; ++i) {
            v4f x = *(v4f*)(dst + 4 * i);
            x *= s;
            *(v4f*)(dst + 4 * i) = x;
        }
    }
    __syncthreads();

    // ---- WMMA compute: each wave -> 2 stacked 16x16 tiles in one 16-col strip.
    const int lane = tid & 31;
    const int wave = tid >> 5;
    const int nt   = wave & 3;
    const int mt0  = (wave >> 2) * 2;

    const int hi = lane >> 4;   // lane half selects K pair {2hi, 2hi+1}
    const int lm = lane & 15;

    v8f acc0 = {};
    v8f acc1 = {};

    const float* Arow0 = &As[(mt0 * 16 + lm) * LDA];
    const float* Arow1 = &As[(mt0 * 16 + 16 + lm) * LDA];
    const int    bcol  = nt * 16 + lm;

#pragma unroll 4
    for (int k = 0; k < C_; k += 4) {
        const int kk = k + 2 * hi;
        // A frag: v0 = A[m][kk], v1 = A[m][kk+1]  (contiguous -> ds_load_b64)
        const v2f a0 = *(const v2f*)(Arow0 + kk);
        const v2f a1 = *(const v2f*)(Arow1 + kk);
        // B frag: v0 = B[kk][n], v1 = B[kk+1][n]
        v2f b;
        b.x = Bs[kk * LDB + bcol];
        b.y = Bs[(kk + 1) * LDB + bcol];
        // D = A x B + C   (8 args: neg_a, A, neg_b, B, c_mod, C, reuse_a, reuse_b)
        acc0 = __builtin_amdgcn_wmma_f32_16x16x4_f32(
                   false, a0, false, b, (short)0, acc0, false, false);
        acc1 = __builtin_amdgcn_wmma_f32_16x16x4_f32(
                   false, a1, false, b, (short)0, acc1, false, false);
    }

    // ---- Epilogue: scores = -(1 - cos)^2 ; store.
    // D layout: VGPR v at lane -> row (v + 8*hi), col lm (within 16x16 tile).
    {
        const int n = colBase + nt * 16 + lm;
#pragma unroll
        for (int v = 0; v < 8; ++v) {
            const int m0 = rowBase + mt0 * 16 + v + 8 * hi;
            float d0 = 1.0f - acc0[v];
            out[(size_t)m0 * K_ + n] = -(d0 * d0);
            const int m1 = m0 + 16;
            float d1 = 1.0f - acc1[v];
            out[(size_t)m1 * K_ + n] = -(d1 * d1);
        }
    }
}

// ---------------------------------------------------------------------------
extern "C" void kernel_launch(void* const* d_in, const int* in_sizes, int n_in,
                              void* d_out, int out_size, void* d_ws, size_t ws_size,
                              hipStream_t stream) {
    const float* emb   = (const float*)d_in[0];   // (N, C) f32
    const float* proto = (const float*)d_in[1];   // (K, C) f32
    float*       out   = (float*)d_out;           // (N, K) f32
    float*       pT    = (float*)d_ws;            // (C, K) f32 = 256 KB scratch

    const int N = in_sizes[0] / C_;               // 131072
    const int K = in_sizes[1] / C_;               // 512

    // Normalize prototypes, store transposed (C x K) for coalesced B staging.
    proto_norm_T_kernel<<<K, C_, 0, stream>>>(proto, pT);

    // Fused normalize + GEMM + score epilogue.
    dim3 grid(N / BM, K / BN);
    cos_score_gemm_kernel<<<grid, 256, 0, stream>>>(emb, pT, out);
}